// Quantizer_10445360464223
// MI455X (gfx1250) — compile-verified
//
#include <hip/hip_runtime.h>

typedef __attribute__((ext_vector_type(2))) float v2f;
typedef __attribute__((ext_vector_type(8))) float v8f;

#define KCODES 1024
#define DIM    64
#define LSTR   66            // padded LDS row stride (floats): +2 banks per code row
#define BATCH  32
#define TLEN   4096
#define BT     (BATCH * TLEN)
#define NELEM  (BATCH * DIM * TLEN)

// Column permutation within a codebook row: K-pairs {0,1} -> words 0..31,
// K-pairs {2,3} -> words 32..63. Makes the two lane-halves of the b64 B-fragment
// read hit disjoint 32-bank sets => conflict-free 64-bank LDS request.
__device__ __forceinline__ int cperm(int c) {
  return ((c & 2) ? 32 : 0) + ((c >> 2) << 1) + (c & 1);
}

// One workgroup = 8 waves (256 threads). Each wave owns 32 consecutive t rows
// (two 16-row M tiles sharing every B fragment => half the LDS traffic/FLOP).
// Full codebook staged in LDS via async global->LDS loads (ASYNCcnt path),
// scanned with V_WMMA_F32_16X16X4_F32 (exact fp32, matches the reference).
__global__ __launch_bounds__(256)
void vq_argmin_kernel(const float* __restrict__ x, const float* __restrict__ emb,
                      float* __restrict__ q_out, float* __restrict__ idx_out,
                      float* __restrict__ accum) {
  __shared__ float se[KCODES * LSTR];   // 270336 B
  __shared__ float e2s[KCODES];         // 4096 B
  __shared__ int   sidx[8][32];         // 1024 B

  const int tid  = threadIdx.x;
  const int wave = tid >> 5;
  const int lane = tid & 31;
  const int ml   = lane & 15;  // M row within tile / N column within tile
  const int hi   = lane >> 4;  // lane-half selects K pair {0,1} vs {2,3}

  // ---- stage codebook into LDS with async global->LDS b64 loads ----
  // 16B of emb per lane-iter: floats [c..c+1] -> word (c>>1), [c+2..c+3] -> word 32+(c>>1)
  for (int i = tid * 4; i < KCODES * DIM; i += 256 * 4) {
    const int code = i >> 6;
    const unsigned lds0 = (unsigned)(uintptr_t)&se[code * LSTR + ((i & 63) >> 1)];
    const unsigned lds1 = lds0 + 32 * 4;
    const unsigned long long g0 = (unsigned long long)(uintptr_t)(emb + i);
    asm volatile("global_load_async_to_lds_b64 %0, %1, off" :: "v"(lds0), "v"(g0) : "memory");
    asm volatile("global_load_async_to_lds_b64 %0, %1, off" :: "v"(lds1), "v"(g0 + 8ull) : "memory");
  }
#if __has_builtin(__builtin_amdgcn_s_wait_asynccnt)
  __builtin_amdgcn_s_wait_asynccnt(0);
#else
  asm volatile("s_wait_asynccnt 0x0" ::: "memory");
#endif
  __syncthreads();

  // ---- per-code squared norms (permutation-invariant sum) ----
  for (int code = tid; code < KCODES; code += 256) {
    float s = 0.f;
    #pragma unroll
    for (int w = 0; w < DIM; ++w) { const float v = se[code * LSTR + w]; s += v * v; }
    e2s[code] = s;
  }
  __syncthreads();

  // ---- load two A tiles (2 x 16 rows x 64 K), kept in registers ----
  const int row_block = blockIdx.x * 256;
  const int b    = row_block >> 12;                  // T = 4096
  const int t0   = (row_block & (TLEN - 1)) + wave * 32;
  const int row0 = row_block + wave * 32;

  v2f a[2][16];
  #pragma unroll
  for (int u = 0; u < 2; ++u) {
    #pragma unroll
    for (int kk = 0; kk < 16; ++kk) {
      const int c0 = kk * 4 + 2 * hi;                // VGPR0 = K{0,2}, VGPR1 = K{1,3}
      const size_t base = ((size_t)(b * DIM + c0)) * TLEN + (size_t)(t0 + u * 16) + ml;
      a[u][kk].x = x[base];
      a[u][kk].y = x[base + TLEN];
    }
  }

  float bestd[2][8];
  int   bestn[2][8];
  #pragma unroll
  for (int u = 0; u < 2; ++u)
    #pragma unroll
    for (int r = 0; r < 8; ++r) { bestd[u][r] = __builtin_inff(); bestn[u][r] = 0; }

  // ---- scan all codes: metric d' = ||e||^2 - 2 x.e  (row ||x||^2 dropped) ----
  for (int n0 = 0; n0 < KCODES; n0 += 16) {
    v8f acc0 = {}, acc1 = {};
    #pragma unroll
    for (int kk = 0; kk < 16; ++kk) {
      // permuted layout: K-pair words = hi*32 + kk*2 (+0,+1)
      const float* bp = &se[(n0 + ml) * LSTR + hi * 32 + kk * 2];
      v2f bf;
      bf.x = bp[0]; bf.y = bp[1];
      acc0 = __builtin_amdgcn_wmma_f32_16x16x4_f32(false, a[0][kk], false, bf,
                                                   (short)0, acc0, false, false);
      acc1 = __builtin_amdgcn_wmma_f32_16x16x4_f32(false, a[1][kk], false, bf,
                                                   (short)0, acc1, false, false);
    }
    const int   n  = n0 + ml;                        // C/D: N = lane&15
    const float en = e2s[n];
    #pragma unroll
    for (int r = 0; r < 8; ++r) {                    // C/D: M = r + 8*hi
      const float d0 = en - 2.0f * acc0[r];
      if (d0 < bestd[0][r]) { bestd[0][r] = d0; bestn[0][r] = n; }
      const float d1 = en - 2.0f * acc1[r];
      if (d1 < bestd[1][r]) { bestd[1][r] = d1; bestn[1][r] = n; }
    }
  }

  // ---- argmin across the 16 lanes of each half (xor keeps halves separate) ----
  #pragma unroll
  for (int u = 0; u < 2; ++u) {
    #pragma unroll
    for (int r = 0; r < 8; ++r) {
      #pragma unroll
      for (int off = 1; off < 16; off <<= 1) {
        const float od = __shfl_xor(bestd[u][r], off, 32);
        const int   on = __shfl_xor(bestn[u][r], off, 32);
        if (od < bestd[u][r] || (od == bestd[u][r] && on < bestn[u][r])) {
          bestd[u][r] = od; bestn[u][r] = on;
        }
      }
    }
  }
  if (ml == 0) {
    #pragma unroll
    for (int u = 0; u < 2; ++u)
      #pragma unroll
      for (int r = 0; r < 8; ++r) {
        const int m = u * 16 + r + 8 * hi;
        sidx[wave][m] = bestn[u][r];
        idx_out[row0 + m] = (float)bestn[u][r];
      }
  }
  __syncthreads();

  // ---- gather q from LDS (inverse perm), coalesced write, fused MSE partial ----
  float lsum = 0.f;
  #pragma unroll
  for (int u = 0; u < 2; ++u) {
    const int m = u * 16 + ml;                       // lane handles rows ml, 16+ml
    const int code = sidx[wave][m];
    #pragma unroll
    for (int cc = 0; cc < 32; ++cc) {
      const int c = hi * 32 + cc;                    // halves split the channel dim
      const float qv = se[code * LSTR + cperm(c)];
      const size_t off = ((size_t)(b * DIM + c)) * TLEN + (size_t)(t0 + m);
      const float xv = x[off];
      q_out[off] = qv;                               // straight-through fwd == q
      const float dd = qv - xv;
      lsum += dd * dd;
    }
  }
  #pragma unroll
  for (int off = 1; off < 32; off <<= 1) lsum += __shfl_xor(lsum, off, 32);
  if (lane == 0) atomicAdd(accum, lsum);
}

__global__ void vq_finalize_kernel(const float* __restrict__ accum,
                                   float* __restrict__ loss_out) {
  const float mse = accum[0] * (1.0f / (float)NELEM);
  loss_out[0] = mse;            // codebook_loss  = mean((q-x)^2)
  loss_out[1] = 0.25f * mse;    // commitment_loss = BETA * mean((q-x)^2)
}

extern "C" void kernel_launch(void* const* d_in, const int* in_sizes, int n_in,
                              void* d_out, int out_size, void* d_ws, size_t ws_size,
                              hipStream_t stream) {
  (void)in_sizes; (void)n_in; (void)out_size; (void)ws_size;
  const float* x   = (const float*)d_in[0];   // [32, 64, 4096] f32
  const float* emb = (const float*)d_in[1];   // [1024, 64] f32

  float* out      = (float*)d_out;
  float* q_out    = out;                              // NELEM floats
  float* loss_out = out + (size_t)NELEM;              // 2 floats
  float* idx_out  = loss_out + 2;                     // BT floats
  float* accum    = (float*)d_ws;

  hipMemsetAsync(d_ws, 0, sizeof(float), stream);

  dim3 grid(BT / 256), block(256);
  vq_argmin_kernel<<<grid, block, 0, stream>>>(x, emb, q_out, idx_out, accum);
  vq_finalize_kernel<<<1, 1, 0, stream>>>(accum, loss_out);
}